// SparseConv_73495480369568
// MI455X (gfx1250) — compile-verified
//
#include <hip/hip_runtime.h>

typedef float v2f __attribute__((ext_vector_type(2)));
typedef float v8f __attribute__((ext_vector_type(8)));

#define CIN        64
#define COUT       64
#define MTOT       65536
#define KOFF       27
#define LDS_STRIDE 72   // 2*72 mod 64 == 16 -> low/high half-wave B reads hit disjoint banks

__global__ void __launch_bounds__(256) zero_out_kernel(float* __restrict__ out, int n) {
  int i = blockIdx.x * blockDim.x + threadIdx.x;
  int stride = gridDim.x * blockDim.x;
  for (; i < n; i += stride) out[i] = 0.0f;
}

__global__ void __launch_bounds__(256) sparse_conv_kernel(
    const float* __restrict__ x,        // [N, 64]
    const float* __restrict__ W,        // [27, 64, 64]
    const int*   __restrict__ in_map,   // [27, M]
    const int*   __restrict__ out_map,  // [27, M]
    float*       __restrict__ out) {    // [N, 64]
  __shared__ float sW[CIN * LDS_STRIDE];

  const int k   = blockIdx.y;
  const int tid = threadIdx.x;

  // Stage W[k] (64x64 f32, 16KB) into LDS, padded row stride 72 floats.
  const float* Wk = W + (size_t)k * CIN * COUT;
  for (int i = tid; i < (CIN * COUT) / 4; i += 256) {
    int row  = i >> 4;            // 16 float4 per row
    int col4 = (i & 15) << 2;
    float4 v = *(const float4*)(Wk + row * COUT + col4);
    *(float4*)&sW[row * LDS_STRIDE + col4] = v;
  }
  __syncthreads();

  const int wave  = tid >> 5;
  const int lane  = tid & 31;
  const int lhalf = lane & 15;
  const int hi    = lane >> 4;    // 0: lanes 0-15, 1: lanes 16-31

  const int* imap = in_map  + (size_t)k * MTOT;
  const int* omap = out_map + (size_t)k * MTOT;

  // 128 blocks * 8 waves * 4 tiles * 16 rows = 65536 = M
  const int tile_base = (blockIdx.x * 8 + wave) * 4;

  for (int t = 0; t < 4; ++t) {
    const int m0 = (tile_base + t) * 16;

    // A-fragment source: lane l<16 holds row m0+l cols {K,K+1}; lane l>=16 holds
    // row m0+(l-16) cols {K+2,K+3}  (ISA 32-bit A 16x4 layout).
    // 32-bit offset + SGPR base: x spans 64MB, offsets fit in u32 -> GVS addressing.
    const unsigned int in_row = (unsigned int)imap[m0 + lhalf];
    const unsigned int xoff   = in_row * (unsigned int)CIN + (unsigned int)(hi * 2);

    v8f acc[4];
#pragma unroll
    for (int cb = 0; cb < 4; ++cb) acc[cb] = (v8f){};

#pragma unroll
    for (int kk = 0; kk < 16; ++kk) {
      // 16x4 fp32 A fragment via per-lane 8-byte gather (saddr + 32-bit voffset)
      v2f a = *(const v2f*)(x + (size_t)(xoff + (unsigned int)(kk * 4)));
      // B fragment base: K-rows kk*4 + {0,1} (low half) / {2,3} (high half)
      const float* brow = &sW[(kk * 4 + hi * 2) * LDS_STRIDE + lhalf];
#pragma unroll
      for (int cb = 0; cb < 4; ++cb) {
        v2f b;
        b.x = brow[cb * 16];               // B[K+0 or K+2][n]
        b.y = brow[cb * 16 + LDS_STRIDE];  // B[K+1 or K+3][n]
        acc[cb] = __builtin_amdgcn_wmma_f32_16x16x4_f32(
            /*neg_a=*/false, a, /*neg_b=*/false, b,
            /*c_mod=*/(short)0, acc[cb], /*reuse_a=*/false, /*reuse_b=*/false);
      }
    }

    // Scatter-add. C/D layout: VGPR g holds M = g + 8*hi, N = lane&15.
    unsigned int obase[8];
#pragma unroll
    for (int g = 0; g < 8; ++g)
      obase[g] = (unsigned int)omap[m0 + hi * 8 + g] * (unsigned int)COUT + (unsigned int)lhalf;

#pragma unroll
    for (int cb = 0; cb < 4; ++cb) {
#pragma unroll
      for (int g = 0; g < 8; ++g) {
        unsafeAtomicAdd((float*)(out + (size_t)(obase[g] + (unsigned int)(cb * 16))), acc[cb][g]);
      }
    }
  }
}

extern "C" void kernel_launch(void* const* d_in, const int* in_sizes, int n_in,
                              void* d_out, int out_size, void* d_ws, size_t ws_size,
                              hipStream_t stream) {
  const float* x       = (const float*)d_in[0];
  const float* W       = (const float*)d_in[1];
  const int*   in_map  = (const int*)d_in[2];
  const int*   out_map = (const int*)d_in[3];
  float*       out     = (float*)d_out;

  zero_out_kernel<<<2048, 256, 0, stream>>>(out, out_size);

  dim3 grid(128, KOFF);
  sparse_conv_kernel<<<grid, 256, 0, stream>>>(x, W, in_map, out_map, out);
}